// MultiVectorEncoder_63153199120926
// MI455X (gfx1250) — compile-verified
//
#include <hip/hip_runtime.h>
#include <math.h>

#define BATCH  64
#define NREG   4
#define NCUE   5                    // 1 cls + 4 registers
#define DIM    768
#define HP     37
#define WP     37
#define NPATCH (HP * WP)            // 1369
#define NTILE  ((NPATCH + 15) / 16) // 86
#define SPLIT  4                    // y-blocks per batch in the sim kernel
#define WAVES_PER_BLOCK 8
#define CHUNK  64                   // K floats per TDM chunk (256B per row)
#define KCHUNKS (DIM / CHUNK)       // 12

typedef __attribute__((ext_vector_type(2))) float        v2f;
typedef __attribute__((ext_vector_type(8))) float        v8f;
typedef __attribute__((ext_vector_type(4))) unsigned int u32x4;
typedef __attribute__((ext_vector_type(8))) int          i32x8;
typedef __attribute__((ext_vector_type(4))) int          i32x4;

// Order-preserving mapping of float -> uint32 (for packed atomicMax argmax).
__device__ __forceinline__ unsigned int orderKey(float v) {
  unsigned int u = __float_as_uint(v);
  return (u & 0x80000000u) ? ~u : (u | 0x80000000u);
}

__global__ void init_ws_kernel(unsigned long long* ws) {
  int i = threadIdx.x;
  if (i < BATCH * NCUE) ws[i] = 0ull;
}

// Issue one TDM transfer: a [rows x CHUNK] fp32 tile (row stride DIM floats)
// from global memory into LDS at byte offset ldsoff. Rows beyond `rows`
// zero-fill via tensor_dim1 OOB semantics. D# packed per CDNA5 ISA §8.
__device__ __forceinline__ void tdm_load_chunk(const float* gaddr,
                                               unsigned ldsoff, int rows) {
  unsigned long long ga = (unsigned long long)(uintptr_t)gaddr;
  u32x4 g0;
  g0[0] = 1u;                                    // count=1, user descriptor
  g0[1] = ldsoff;                                // lds_addr (bytes)
  g0[2] = (unsigned)(ga & 0xffffffffu);          // global_addr[31:0]
  g0[3] = (unsigned)((ga >> 32) & 0x01ffffffu)   // global_addr[56:32]
          | (2u << 30);                          // type = 2 ("image")
  i32x8 g1;
  g1[0] = 0x00020000;                            // data_size = 2 (4 bytes)
  g1[1] = (int)((DIM & 0xffffu) << 16);          // tensor_dim0[15:0] = 768
  g1[2] = (int)(((unsigned)(DIM >> 16) & 0xffffu) |
                (((unsigned)rows & 0xffffu) << 16));  // dim0 hi | tensor_dim1 lo
  g1[3] = (int)((((unsigned)rows >> 16) & 0xffffu) |
                ((unsigned)CHUNK << 16));        // tensor_dim1 hi | tile_dim0
  g1[4] = 16;                                    // tile_dim1=16, tile_dim2=0
  g1[5] = DIM;                                   // tensor_dim0_stride[31:0]
  g1[6] = 0;                                     // stride hi | dim1_stride lo
  g1[7] = 0;
  i32x4 gz4 = {0, 0, 0, 0};                      // groups 2/3 unused (2-D tile)
  i32x8 gz8 = {0, 0, 0, 0, 0, 0, 0, 0};
  // clang-23 / therock-10.0 form: 6 args (extra i32x8 group), cpol last.
  __builtin_amdgcn_tensor_load_to_lds(g0, g1, gz4, gz4, gz8, 0);
}

// Kernel 1: sim = cues . patches^T via V_WMMA_F32_16X16X4_F32 with the
// A-matrix (16 patches x K) staged in LDS by the Tensor Data Mover,
// double-buffered in K-chunks; fused per-cue argmax via packed atomicMax.
// D[16,16] layout: lane L holds cue (L%16), patches tile*16 + 8*(L/16) + j.
__global__ __launch_bounds__(256) void sim_argmax_kernel(
    const float* __restrict__ cls_tok,
    const float* __restrict__ regs,
    const float* __restrict__ patches,
    unsigned long long* __restrict__ wsmax) {
  __shared__ float lds[WAVES_PER_BLOCK][2][16 * CHUNK];   // 64 KB

  const int b      = blockIdx.x;
  const int lane   = threadIdx.x & 31;
  const int wave   = threadIdx.x >> 5;
  const int wgWave = blockIdx.y * WAVES_PER_BLOCK + wave;  // 0..31
  const int nwaves = gridDim.y * WAVES_PER_BLOCK;          // 32

  const int n    = lane & 15;   // B column (cue) / A row-in-tile for loads
  const int half = lane >> 4;

  const float* __restrict__ pbase = patches + (size_t)b * NPATCH * DIM;
  // n==0 and n>=NCUE lanes read the cls row; columns n>=NCUE are never used.
  const float* cue = (n >= 1 && n < NCUE)
                         ? regs + ((size_t)b * NREG + (n - 1)) * DIM
                         : cls_tok + (size_t)b * DIM;

  float* buf0 = &lds[wave][0][0];
  float* buf1 = &lds[wave][1][0];
  const unsigned off0 = (unsigned)(uintptr_t)buf0;  // LDS byte offset
  const unsigned off1 = (unsigned)(uintptr_t)buf1;

  const int ntiles = (NTILE - 1 - wgWave) / nwaves + 1;  // wgWave < NTILE
  const int total  = ntiles * KCHUNKS;

  float bestv = -__builtin_inff();
  int   besti = 0;

  // Issue TDM for pipeline chunk i (flattened over (tile, kchunk)).
  auto issue = [&](int i) {
    const int t  = wgWave + (i / KCHUNKS) * nwaves;
    const int kc = (i % KCHUNKS) * CHUNK;
    const int p0 = t * 16;
    int rows = NPATCH - p0;
    if (rows > 16) rows = 16;
    tdm_load_chunk(pbase + (size_t)p0 * DIM + kc, (i & 1) ? off1 : off0, rows);
  };

  issue(0);
  if (total > 1) issue(1);

  v8f acc;
#pragma unroll
  for (int j = 0; j < 8; ++j) acc[j] = 0.f;

  for (int i = 0; i < total; ++i) {
    // Chunks complete in order: with <=2 outstanding, tensorcnt<=1 (or 0 at
    // the tail) guarantees chunk i has landed in LDS.
    if (i + 1 < total) __builtin_amdgcn_s_wait_tensorcnt(1);
    else               __builtin_amdgcn_s_wait_tensorcnt(0);
    asm volatile("" ::: "memory");   // keep LDS reads below the wait

    const float* __restrict__ bufc = (i & 1) ? buf1 : buf0;
    const float* __restrict__ arow = bufc + n * CHUNK;
    const int kc = (i % KCHUNKS) * CHUNK;

#pragma unroll 8
    for (int lk = 0; lk < CHUNK; lk += 4) {
      const int ks = lk + 2 * half;                 // even -> 8B aligned
      v2f a  = *(const v2f*)(arow + ks);            // A[16,4] frag from LDS
      v2f bf = *(const v2f*)(cue + kc + ks);        // B[4,16] frag (cached)
      acc = __builtin_amdgcn_wmma_f32_16x16x4_f32(
          /*neg_a=*/false, a, /*neg_b=*/false, bf,
          /*c_mod=*/(short)0, acc, /*reuse_a=*/false, /*reuse_b=*/false);
    }

    if ((i % KCHUNKS) == KCHUNKS - 1) {             // tile finished: readout
      const int t = wgWave + (i / KCHUNKS) * nwaves;
      if (n < NCUE) {
#pragma unroll
        for (int j = 0; j < 8; ++j) {
          const int   p = t * 16 + 8 * half + j;
          const float v = acc[j];
          if (p < NPATCH && v > bestv) { bestv = v; besti = p; }
        }
      }
#pragma unroll
      for (int j = 0; j < 8; ++j) acc[j] = 0.f;
    }

    if (i + 2 < total) {
      // Buffer (i&1) was just read; ensure our ds_loads drained before the
      // TDM engine overwrites it.
      asm volatile("s_wait_dscnt 0" ::: "memory");
      issue(i + 2);
    }
  }

  // Lanes L and L^16 hold the same cue column: merge their patch halves.
  {
    float ov = __shfl_xor(bestv, 16, 32);
    int   oi = __shfl_xor(besti, 16, 32);
    if (ov > bestv || (ov == bestv && oi < besti)) { bestv = ov; besti = oi; }
  }
  if (half == 0 && n < NCUE) {
    // ~besti in low bits: on equal sim, max picks the SMALLER index
    // (matches jnp.argmax first-occurrence semantics).
    unsigned long long packed =
        ((unsigned long long)orderKey(bestv) << 32) | (unsigned int)(~besti);
    atomicMax(&wsmax[b * NCUE + n], packed);
  }
}

// Kernel 2: ROI window mean + L2 normalization of all 10 tokens per batch.
__global__ __launch_bounds__(256) void finalize_kernel(
    const float* __restrict__ cls_tok,
    const float* __restrict__ regs,
    const float* __restrict__ patches,
    const int* __restrict__ roi_side_p,
    const unsigned long long* __restrict__ wsmax,
    float* __restrict__ out) {
  const int b   = blockIdx.x;
  const int tt  = blockIdx.y;   // 0..4 cues, 5..9 ROI tokens
  const int tid = threadIdx.x;
  __shared__ float red[256];

  float vals[DIM / 256];   // 3 elements per thread
  if (tt < NCUE) {
    const float* src = (tt == 0) ? cls_tok + (size_t)b * DIM
                                 : regs + ((size_t)b * NREG + (tt - 1)) * DIM;
#pragma unroll
    for (int i = 0; i < DIM / 256; ++i) vals[i] = src[tid + 256 * i];
  } else {
    const int c = tt - NCUE;
    const unsigned long long packed = wsmax[b * NCUE + c];
    const int idx = (int)(~(unsigned int)(packed & 0xffffffffu));
    const int hh = idx / WP, ww = idx % WP;
    const int r  = roi_side_p[0] / 2;
    const int h0 = max(0, hh - r), h1 = min(HP - 1, hh + r);
    const int w0 = max(0, ww - r), w1 = min(WP - 1, ww + r);
    const float inv = 1.0f / (float)((h1 - h0 + 1) * (w1 - w0 + 1));
    float s[DIM / 256] = {0.f, 0.f, 0.f};
    for (int h = h0; h <= h1; ++h) {
      for (int w = w0; w <= w1; ++w) {
        const float* prow = patches + ((size_t)b * NPATCH + h * WP + w) * DIM;
#pragma unroll
        for (int i = 0; i < DIM / 256; ++i) s[i] += prow[tid + 256 * i];
      }
    }
#pragma unroll
    for (int i = 0; i < DIM / 256; ++i) vals[i] = s[i] * inv;
  }

  float sq = 0.f;
#pragma unroll
  for (int i = 0; i < DIM / 256; ++i) sq += vals[i] * vals[i];
  red[tid] = sq;
  __syncthreads();
  for (int s = 128; s > 0; s >>= 1) {
    if (tid < s) red[tid] += red[tid + s];
    __syncthreads();
  }
  const float invn = 1.0f / fmaxf(sqrtf(red[0]), 1e-12f);

  float* dst = out + ((size_t)b * 2 * NCUE + tt) * DIM;
#pragma unroll
  for (int i = 0; i < DIM / 256; ++i) dst[tid + 256 * i] = vals[i] * invn;
}

extern "C" void kernel_launch(void* const* d_in, const int* in_sizes, int n_in,
                              void* d_out, int out_size, void* d_ws, size_t ws_size,
                              hipStream_t stream) {
  const float* cls     = (const float*)d_in[0];
  const float* regs    = (const float*)d_in[1];
  const float* patches = (const float*)d_in[2];
  const int*   roi     = (const int*)d_in[3];
  float* out = (float*)d_out;
  unsigned long long* ws = (unsigned long long*)d_ws;

  init_ws_kernel<<<1, BATCH * NCUE, 0, stream>>>(ws);
  sim_argmax_kernel<<<dim3(BATCH, SPLIT), 256, 0, stream>>>(cls, regs, patches, ws);
  finalize_kernel<<<dim3(BATCH, 2 * NCUE), 256, 0, stream>>>(cls, regs, patches,
                                                             roi, ws, out);
}